// DCTFeatureModel_50783693308121
// MI455X (gfx1250) — compile-verified
//
#include <hip/hip_runtime.h>

typedef __attribute__((ext_vector_type(2))) float v2f;
typedef __attribute__((ext_vector_type(4))) float v4f;
typedef __attribute__((ext_vector_type(8))) float v8f;

#define BATCH     1024
#define NSW       2
#define DCT_NBINS 8
#define NDCT      32
#define NF        64
#define HW        8
#define KDIM      2048          // NDCT * HW * HW
#define SLOPE     0.02f

// ---------------------------------------------------------------------------
// Kernel 1: fold the 3D DCT into the conv weights.
// V[so][n*64+i*8+j] = sum_{f,p,q} Ct[f,n] * Cs[p,i] * Cs[q,j] * W[so][f*64+p*8+q]
// One block per (s,o) pair (128 blocks), 256 threads, separable passes in LDS.
// ---------------------------------------------------------------------------
__global__ __launch_bounds__(256) void dct_build_v(const float* __restrict__ W,
                                                   float* __restrict__ V) {
    __shared__ float bufA[KDIM];
    __shared__ float bufB[KDIM];
    __shared__ float Ct[NDCT * NDCT];
    __shared__ float Cs[HW * HW];

    const int tid = threadIdx.x;
    const int so  = blockIdx.x;           // 0..127  == s*64 + o
    const float PI = 3.14159265358979323846f;

    for (int e = tid; e < NDCT * NDCT; e += 256) {
        int f = e >> 5, n = e & 31;
        Ct[e] = 2.0f * __cosf(PI * (2.0f * n + 1.0f) * f / (2.0f * NDCT));
    }
    if (tid < HW * HW) {
        int p = tid >> 3, i = tid & 7;
        Cs[tid] = 2.0f * __cosf(PI * (2.0f * i + 1.0f) * p / (2.0f * HW));
    }
    for (int e = tid; e < KDIM; e += 256)
        bufA[e] = W[(size_t)so * KDIM + e];
    __syncthreads();

    // pass 1: contract f with Ct  ->  bufB[n*64 + pq]
    for (int e = tid; e < KDIM; e += 256) {
        int n = e >> 6, pq = e & 63;
        float acc = 0.f;
        #pragma unroll
        for (int f = 0; f < NDCT; ++f) acc += Ct[f * 32 + n] * bufA[f * 64 + pq];
        bufB[e] = acc;
    }
    __syncthreads();

    // pass 2: contract p with Cs  ->  bufA[n*64 + i*8 + q]
    for (int e = tid; e < KDIM; e += 256) {
        int n = e >> 6, i = (e >> 3) & 7, q = e & 7;
        float acc = 0.f;
        #pragma unroll
        for (int p = 0; p < HW; ++p) acc += Cs[p * 8 + i] * bufB[n * 64 + p * 8 + q];
        bufA[e] = acc;
    }
    __syncthreads();

    // pass 3: contract q with Cs  ->  V[so][n*64 + i*8 + j]
    for (int e = tid; e < KDIM; e += 256) {
        int ni = e >> 3, j = e & 7;
        float acc = 0.f;
        #pragma unroll
        for (int q = 0; q < HW; ++q) acc += Cs[q * 8 + j] * bufA[ni * 8 + q];
        V[(size_t)so * KDIM + e] = acc;
    }
}

// ---------------------------------------------------------------------------
// Kernel 2 (bandwidth-critical): mean over the 8 DCT bins.
// xr[s][b][k] = 0.125 * sum_c x[b*32768 + s*16384 + c*2048 + k],  k in [0,2048)
// 128-bit loads/stores; streams x (134 MB) exactly once.
// ---------------------------------------------------------------------------
__global__ __launch_bounds__(256) void reduce_bins(const float* __restrict__ x,
                                                   float* __restrict__ xr) {
    const int gid = blockIdx.x * 256 + threadIdx.x;      // 0 .. 2^20-1 (float4 units)
    const int s   = gid >> 19;
    const int b   = (gid >> 9) & 1023;
    const int k4  = (gid & 511) << 2;                    // k aligned to 4

    const v4f* src = (const v4f*)(x + (size_t)b * 32768 + (size_t)s * 16384 + k4);
    v4f acc = src[0];
    #pragma unroll
    for (int c = 1; c < DCT_NBINS; ++c) acc += src[c * 512];   // 2048 floats / 4
    acc *= 0.125f;
    ((v4f*)xr)[gid] = acc;
}

// ---------------------------------------------------------------------------
// Kernel 3: feat[b, s*64+o] = leaky( xr[s][b][:] . V[s*64+o][:] + bias[s*64+o] )
// One wave per 16x16 output tile, K=2048 via v_wmma_f32_16x16x4_f32.
// A layout (16x4 f32): lanes 0-15 row=lane K={k0,k0+1}; lanes 16-31 row=lane-16
// K={k0+2,k0+3}. B layout mirrors with column=lane&15. C/D: N=lane&15,
// M = vgpr + 8*(lane>=16).
// ---------------------------------------------------------------------------
__global__ __launch_bounds__(128) void gemm_wmma(const float* __restrict__ xr,
                                                 const float* __restrict__ V,
                                                 const float* __restrict__ bias,
                                                 float* __restrict__ out) {
    const int wave = blockIdx.x * (blockDim.x >> 5) + (threadIdx.x >> 5); // 0..511
    const int lane = threadIdx.x & 31;

    const int s   = wave >> 8;          // 0..1
    const int rem = wave & 255;
    const int b0  = (rem >> 2) << 4;    // batch tile base (64 tiles)
    const int o0  = (rem & 3) << 4;     // output-filter tile base (4 tiles)

    const int half = lane >> 4;         // K sub-offset group
    const int l15  = lane & 15;

    const float* aptr = xr + (size_t)s * (BATCH * KDIM) + (size_t)(b0 + l15) * KDIM + 2 * half;
    const float* bptr = V  + (size_t)(s * NF + o0 + l15) * KDIM + 2 * half;

    v8f c = {0.f, 0.f, 0.f, 0.f, 0.f, 0.f, 0.f, 0.f};

    #pragma unroll 8
    for (int k0 = 0; k0 < KDIM; k0 += 4) {
        v2f a  = *(const v2f*)aptr;
        v2f bb = *(const v2f*)bptr;
        c = __builtin_amdgcn_wmma_f32_16x16x4_f32(false, a, false, bb,
                                                  (short)0, c, false, false);
        aptr += 4;
        bptr += 4;
    }

    const float bv  = bias[s * NF + o0 + l15];
    const int   col = s * NF + o0 + l15;
    #pragma unroll
    for (int r = 0; r < 8; ++r) {
        int   row = b0 + r + 8 * half;
        float v   = c[r] + bv;
        v = (v >= 0.f) ? v : SLOPE * v;
        out[(size_t)row * (NSW * NF) + col] = v;
    }
}

// ---------------------------------------------------------------------------
extern "C" void kernel_launch(void* const* d_in, const int* in_sizes, int n_in,
                              void* d_out, int out_size, void* d_ws, size_t ws_size,
                              hipStream_t stream) {
    const float* x    = (const float*)d_in[0];   // (1024, 1, 512, 8, 8)
    const float* W    = (const float*)d_in[1];   // (2, 64, 32, 8, 8)
    const float* bias = (const float*)d_in[2];   // (2, 64)
    float*       out  = (float*)d_out;           // (1024, 128)

    float* xr = (float*)d_ws;                    // 2*1024*2048 floats = 16 MB
    float* V  = xr + (size_t)NSW * BATCH * KDIM; // 2*64*2048 floats  = 1 MB

    dct_build_v<<<NSW * NF, 256, 0, stream>>>(W, V);
    reduce_bins<<<(NSW * BATCH * KDIM / 4) / 256, 256, 0, stream>>>(x, xr);
    gemm_wmma<<<512 / 4, 128, 0, stream>>>(xr, V, bias, out);
}